// DSVF_49649821941763
// MI455X (gfx1250) — compile-verified
//
#include <hip/hip_runtime.h>
#include <math.h>

typedef float v2f __attribute__((ext_vector_type(2)));
typedef float v8f __attribute__((ext_vector_type(8)));

constexpr int T_LEN = 8192;   // time length per row (reference: T=8192)

// One wave (32 lanes) processes a tile of 16 rows across all of T in chunks
// of 16 time steps. Per chunk:
//   Y(16x16) = U(16x16) x H^T  via 4x v_wmma_f32_16x16x4_f32 (K=16 split in 4)
// plus per-row state-carry terms p_j*y[-1] + q_j*y[-2].
__global__ __launch_bounds__(32) void dsvf_wmma_kernel(
    const float* __restrict__ x,
    const float* __restrict__ gp,
    const float* __restrict__ rp,
    const float* __restrict__ mhp_p,
    const float* __restrict__ mbp_p,
    const float* __restrict__ mlp_p,
    float* __restrict__ yout)
{
    __shared__ float xs[16][17];   // x tile (padded vs bank conflicts)
    __shared__ float us[16][17];   // FIR output tile
    __shared__ float cx1[16], cx2[16];   // x[t-1], x[t-2] per row
    __shared__ float cy1[16], cy2[16];   // y[t-1], y[t-2] per row
    __shared__ float hs[20];             // impulse response h_0..h_16

    const unsigned lane = threadIdx.x;   // 0..31 (wave32)
    const unsigned j    = lane & 15u;    // output column (time within chunk)
    const bool     hi   = lane >= 16u;   // upper half-wave
    const int rbase = blockIdx.x * 16;

    // ---- coefficient computation (identical math to reference) ----
    const float gv  = gp[0];
    const float rv  = rp[0];
    const float mhp = mhp_p[0];
    const float mbp = mbp_p[0];
    const float mlp = mlp_p[0];

    const float sig = 1.0f / (1.0f + expf(-gv));
    const float gg  = tanf(3.14159265358979323846f * 0.5f * sig);
    const float rr  = log1pf(expf(rv));          // softplus
    const float g2  = gg * gg;

    // b is NOT normalized in the reference (divided by a[0]==1 post-normalize)
    const float b0 = g2 * mlp + gg * mbp + mhp;
    const float b1 = 2.0f * g2 * mlp - 2.0f * mhp;
    const float b2 = g2 * mlp - gg * mbp + mhp;
    const float a0 = g2 + 2.0f * rr * gg + 1.0f;
    const float inva0 = 1.0f / a0;
    const float a1 = (2.0f * g2 - 2.0f) * inva0;
    const float a2 = (g2 - 2.0f * rr * gg + 1.0f) * inva0;

    // impulse response of 1/(1 + a1 z^-1 + a2 z^-2), h_0..h_16
    if (lane == 0u) {
        hs[0] = 1.0f;
        hs[1] = -a1;
        for (int i = 2; i <= 16; ++i)
            hs[i] = -a1 * hs[i - 1] - a2 * hs[i - 2];
    }
    if (lane < 16u) {
        cx1[lane] = 0.0f; cx2[lane] = 0.0f;
        cy1[lane] = 0.0f; cy2[lane] = 0.0f;
    }
    __syncthreads();

    // ---- constant B fragments: B[k][n] = h_{n-k} (upper-tri Toeplitz) ----
    // B 4x16 layout per WMMA: VGPR0 lanes0-15=K0,N=lane / lanes16-31=K2 ;
    //                         VGPR1 lanes0-15=K1        / lanes16-31=K3
    v2f Bf[4];
#pragma unroll
    for (int s = 0; s < 4; ++s) {
        const int k0 = 4 * s + (hi ? 2 : 0);
        const int k1 = k0 + 1;
        Bf[s].x = ((int)j >= k0) ? hs[j - k0] : 0.0f;
        Bf[s].y = ((int)j >= k1) ? hs[j - k1] : 0.0f;
    }
    const float pj = hs[j + 1];          // coefficient of y[t0-1]
    const float qj = -a2 * hs[j];        // coefficient of y[t0-2]

    // load assignment: lane -> (row = lane/2, 8 contiguous t at toff)
    const int rload = (int)(lane >> 1);
    const int toff  = (int)(lane & 1u) * 8;
    const float* xrow = x + (size_t)(rbase + rload) * T_LEN;

    for (int t0 = 0; t0 < T_LEN; t0 += 16) {
        // ---- coalesced global load of the 16x16 x tile (2x float4/lane) ----
        const float4 xa = *(const float4*)(xrow + t0 + toff);
        const float4 xb = *(const float4*)(xrow + t0 + toff + 4);
        if (t0 + 16 < T_LEN)
            __builtin_prefetch(xrow + t0 + 16 + toff, 0, 1);  // global_prefetch_b8

        xs[rload][toff + 0] = xa.x; xs[rload][toff + 1] = xa.y;
        xs[rload][toff + 2] = xa.z; xs[rload][toff + 3] = xa.w;
        xs[rload][toff + 4] = xb.x; xs[rload][toff + 5] = xb.y;
        xs[rload][toff + 6] = xb.z; xs[rload][toff + 7] = xb.w;
        __syncthreads();

        // ---- FIR part: u = b0*x_t + b1*x_{t-1} + b2*x_{t-2} ----
        float prev1 = toff ? xs[rload][7] : cx1[rload];
        float prev2 = toff ? xs[rload][6] : cx2[rload];
#pragma unroll
        for (int k = 0; k < 8; ++k) {
            const float xv = xs[rload][toff + k];
            us[rload][toff + k] = b0 * xv + b1 * prev1 + b2 * prev2;
            prev2 = prev1;
            prev1 = xv;
        }
        __syncthreads();

        // update x carries for next chunk (xs of this chunk still live)
        if (toff) {
            cx1[rload] = xs[rload][15];
            cx2[rload] = xs[rload][14];
        }

        // ---- IIR within chunk: Y = U x B via 4 chained WMMA (K=16) ----
        // A 16x4 layout: lanes0-15 M=j hold K0,K1 ; lanes16-31 M=j hold K2,K3
        v8f acc = {};
#pragma unroll
        for (int s = 0; s < 4; ++s) {
            const int k0 = 4 * s + (hi ? 2 : 0);
            v2f A;
            A.x = us[j][k0];
            A.y = us[j][k0 + 1];
            acc = __builtin_amdgcn_wmma_f32_16x16x4_f32(
                /*neg_a=*/false, A, /*neg_b=*/false, Bf[s],
                /*c_mod=*/(short)0, acc, /*reuse_a=*/false, /*reuse_b=*/false);
        }

        // ---- add state-carry terms and store (column-coalesced) ----
        // D layout: VGPR v, lanes0-15 -> row v, N=lane ; lanes16-31 -> row v+8
        float outv[8];
#pragma unroll
        for (int v = 0; v < 8; ++v) {
            const int r2 = v + (hi ? 8 : 0);
            const float yv = acc[v] + pj * cy1[r2] + qj * cy2[r2];
            outv[v] = yv;
            yout[(size_t)(rbase + r2) * T_LEN + t0 + j] = yv;
        }
        __syncthreads();   // all reads of cy1/cy2 done before update

        // lanes holding columns 15 / 14 own the new y carries
        if (j == 15u) {
#pragma unroll
            for (int v = 0; v < 8; ++v) cy1[v + (hi ? 8 : 0)] = outv[v];
        }
        if (j == 14u) {
#pragma unroll
            for (int v = 0; v < 8; ++v) cy2[v + (hi ? 8 : 0)] = outv[v];
        }
        __syncthreads();
    }
}

extern "C" void kernel_launch(void* const* d_in, const int* in_sizes, int n_in,
                              void* d_out, int out_size, void* d_ws, size_t ws_size,
                              hipStream_t stream) {
    const float* x   = (const float*)d_in[0];
    const float* g   = (const float*)d_in[1];
    const float* r   = (const float*)d_in[2];
    const float* mhp = (const float*)d_in[3];
    const float* mbp = (const float*)d_in[4];
    const float* mlp = (const float*)d_in[5];
    float* y = (float*)d_out;

    const int B = in_sizes[0] / T_LEN;          // 2048
    dim3 grid(B / 16), block(32);               // 1 wave / workgroup, 16 rows each
    hipLaunchKernelGGL(dsvf_wmma_kernel, grid, block, 0, stream,
                       x, g, r, mhp, mbp, mlp, y);
}